// VNN_46308337385504
// MI455X (gfx1250) — compile-verified
//
#include <hip/hip_runtime.h>
#include <stdint.h>

// ---------------------------------------------------------------------------
// VNEmbedding multi-scale KNN features for MI455X (gfx1250, wave32).
//
// B=4 batches of N=8192 points in R^3. Per-point top-32 nearest neighbors
// (rank by 2<q,c> - |c|^2; row-constant -|q|^2 dropped: order-invariant),
// then neighbor means for k=8/16/32 (k=16 reproduces the torch flat-reshape
// gather), then the closed-form view-scramble output assembly.
//
// Roofline: HBM traffic trivial (<4 MB); never materialize the 1 GB distance
// matrix. Candidates staged in LDS (48 KB chunks) via one TENSOR_LOAD_TO_LDS
// 2D-tile DMA per chunk (async-to-LDS / plain copy fallbacks). One wave owns
// 16 query rows; two V_WMMA_F32_16X16X4_F32 per 32-candidate step produce the
// 16x32 Gram tile; rows reassembled with v_permlane16_swap_b32 (one VALU op
// per register pair); per-row sorted top-32 kept via a v_readlane-threshold-
// gated wave-wide bitonic merge (noinline, shared across rows).
// ---------------------------------------------------------------------------

#define N_PTS       8192
#define BATCH       4
#define CHUNK       4096        // candidates per LDS refill: 3*4096*4B = 48 KB
#define WG_THREADS  128         // 4 waves; each wave owns 16 query rows
#define ROWS_PER_WG 64
#define ZROW        (3 * CHUNK) // 16 zero floats appended for K=3 padding

typedef __attribute__((ext_vector_type(2))) float        v2f;
typedef __attribute__((ext_vector_type(8))) float        v8f;
typedef __attribute__((ext_vector_type(4))) int          v4i;
typedef __attribute__((ext_vector_type(8))) int          v8i;
typedef __attribute__((ext_vector_type(4))) unsigned int v4u;
typedef __attribute__((ext_vector_type(2))) unsigned int v2u;
typedef __attribute__((address_space(1))) v4i* gv4ip;    // global src
typedef __attribute__((address_space(3))) v4i* lv4ip;    // LDS dst

#if __has_builtin(__builtin_amdgcn_tensor_load_to_lds) && \
    __has_builtin(__builtin_amdgcn_s_wait_tensorcnt)
#define HAVE_TDM 1
#endif
#if __has_builtin(__builtin_amdgcn_global_load_async_to_lds_b128)
#define HAVE_ASYNC_LDS 1
#endif

struct VI { float v; int i; };

__device__ __forceinline__ float readlane_f32(float v, int l) {
  return __int_as_float(__builtin_amdgcn_readlane(__float_as_int(v), l));
}

#ifdef HAVE_ASYNC_LDS
__device__ __forceinline__ void wait_async_zero() {
#if __has_builtin(__builtin_amdgcn_s_wait_asynccnt)
  __builtin_amdgcn_s_wait_asynccnt(0);
#else
  asm volatile("s_wait_asynccnt 0" ::: "memory");
#endif
}
#endif

// Merge resident top-32 (res: one (v,i) per lane, ascending by rank across the
// wave) with 32 new candidates (nw, unsorted).
// rank: larger value first; ties -> smaller index (matches jax.lax.top_k).
__device__ __attribute__((noinline)) VI topk_merge32(VI nw, VI res) {
  const int lane = threadIdx.x & 31;
  // 1) bitonic sort of the new batch, ascending by rank
#pragma unroll
  for (int k = 2; k <= 32; k <<= 1) {
#pragma unroll
    for (int j = k >> 1; j >= 1; j >>= 1) {
      float pv = __shfl_xor(nw.v, j);
      int   pi = __shfl_xor(nw.i, j);
      bool pBetter   = (pv > nw.v) || (pv == nw.v && pi < nw.i);
      bool keepSmall = (((lane & k) == 0) == ((lane & j) == 0));
      bool takeP     = keepSmall ? !pBetter : pBetter;
      nw.v = takeP ? pv : nw.v;
      nw.i = takeP ? pi : nw.i;
    }
  }
  // 2) reverse -> descending; elementwise keep-better vs ascending resident
  float rv = __shfl_xor(nw.v, 31);
  int   ri = __shfl_xor(nw.i, 31);
  bool revBetter = (rv > res.v) || (rv == res.v && ri < res.i);
  VI m;
  m.v = revBetter ? rv : res.v;
  m.i = revBetter ? ri : res.i;
  // 3) bitonic merge of the bitonic sequence -> ascending by rank
#pragma unroll
  for (int j = 16; j >= 1; j >>= 1) {
    float pv = __shfl_xor(m.v, j);
    int   pi = __shfl_xor(m.i, j);
    bool pBetter   = (pv > m.v) || (pv == m.v && pi < m.i);
    bool keepSmall = ((lane & j) == 0);
    bool takeP     = keepSmall ? !pBetter : pBetter;
    m.v = takeP ? pv : m.v;
    m.i = takeP ? pi : m.i;
  }
  return m;
}

// One-shot per-row feature emit: tree-reduce neighbor sums; extract the k=8 /
// k=16 prefixes at group boundaries with v_readlane (resident list ascending:
// lanes 24..31 = top-8, lanes 16..31 = top-16). noinline: one body, 16 calls.
__device__ __attribute__((noinline))
void row_stats(const float* xb, int i, float* m8, float* m16, float* m32,
               int orow) {
  const int lane = threadIdx.x & 31;
  float g0 = xb[0 * N_PTS + i];
  float g1 = xb[1 * N_PTS + i];
  float g2 = xb[2 * N_PTS + i];
  float h0 = xb[3 * i + 0];     // torch flat-reshape gather (3*idx+c < 3N)
  float h1 = xb[3 * i + 1];
  float h2 = xb[3 * i + 2];
#pragma unroll
  for (int d = 1; d <= 4; d <<= 1) {     // 8-lane group sums
    g0 += __shfl_xor(g0, d); g1 += __shfl_xor(g1, d); g2 += __shfl_xor(g2, d);
    h0 += __shfl_xor(h0, d); h1 += __shfl_xor(h1, d); h2 += __shfl_xor(h2, d);
  }
  float s8_0 = readlane_f32(g0, 24);     // top-8 sums live in lanes 24..31
  float s8_1 = readlane_f32(g1, 24);
  float s8_2 = readlane_f32(g2, 24);
  g0 += __shfl_xor(g0, 8); g1 += __shfl_xor(g1, 8); g2 += __shfl_xor(g2, 8);
  h0 += __shfl_xor(h0, 8); h1 += __shfl_xor(h1, 8); h2 += __shfl_xor(h2, 8);
  float s16_0 = readlane_f32(h0, 16);    // top-16 sums live in lanes 16..31
  float s16_1 = readlane_f32(h1, 16);
  float s16_2 = readlane_f32(h2, 16);
  g0 += __shfl_xor(g0, 16); g1 += __shfl_xor(g1, 16); g2 += __shfl_xor(g2, 16);
  if (lane == 0) {
    int o = orow * 3;
    m32[o + 0] = g0 * (1.0f / 32.0f);
    m32[o + 1] = g1 * (1.0f / 32.0f);
    m32[o + 2] = g2 * (1.0f / 32.0f);
    m16[o + 0] = s16_0 * (1.0f / 16.0f);
    m16[o + 1] = s16_1 * (1.0f / 16.0f);
    m16[o + 2] = s16_2 * (1.0f / 16.0f);
    m8[o + 0]  = s8_0 * (1.0f / 8.0f);
    m8[o + 1]  = s8_1 * (1.0f / 8.0f);
    m8[o + 2]  = s8_2 * (1.0f / 8.0f);
  }
}

__global__ void __launch_bounds__(WG_THREADS)
knn_feat_kernel(const float* __restrict__ xg,
                float* __restrict__ m8,
                float* __restrict__ m16,
                float* __restrict__ m32) {
  __shared__ float lds[3 * CHUNK + 16];   // [3][CHUNK] + 16-float zero row

  const int tid  = threadIdx.x;
  const int lane = tid & 31;
  const int wave = tid >> 5;
  const int b       = blockIdx.x / (N_PTS / ROWS_PER_WG);
  const int wgRow   = (blockIdx.x % (N_PTS / ROWS_PER_WG)) * ROWS_PER_WG;
  const int rowBase = wgRow + wave * 16;
  const float* xb = xg + (size_t)b * 3 * N_PTS;   // [3][N] block of batch b

  if (tid < 16) lds[ZROW + tid] = 0.0f;           // K=3 zero padding row

  // ---- WMMA A operand: 16 query points as a 16x4 f32 matrix (K = x,y,z,0).
  // ISA 7.12.2 layout: lanes 0-15 hold K=0 (vgpr0), K=1 (vgpr1);
  //                    lanes 16-31 hold K=2 (vgpr0), K=3 (vgpr1, zero).
  const int qm = lane & 15;
  v2f amat;
  amat.x = xb[((lane < 16) ? 0 : 2) * N_PTS + rowBase + qm];
  amat.y = (lane < 16) ? xb[1 * N_PTS + rowBase + qm] : 0.0f;

  VI res[16];
#pragma unroll
  for (int r = 0; r < 16; ++r) { res[r].v = -__builtin_huge_valf(); res[r].i = 0x7fffffff; }

  for (int ch = 0; ch < N_PTS / CHUNK; ++ch) {
    const int cstart = ch * CHUNK;

    // ---- stage candidate chunk into LDS: a 2D tile (CHUNK x 3) of the
    //      (N_PTS x 3) tensor, row stride N_PTS, packed into LDS [3][CHUNK].
#ifdef HAVE_TDM
    if (tid == 0) {
      uint32_t ldsb  = (uint32_t)(uintptr_t)(lv4ip)(&lds[0]);   // LDS byte addr
      uint64_t gaddr = (uint64_t)(uintptr_t)(xb + cstart);      // tile start
      v4u g0;
      g0[0] = 1u;                                  // count=1, user descriptor
      g0[1] = ldsb;                                // lds_addr
      g0[2] = (uint32_t)(gaddr & 0xffffffffu);     // global_addr[31:0]
      g0[3] = (uint32_t)((gaddr >> 32) & 0x1ffffffu) | (2u << 30); // [56:32]|type=2
      v8i g1;
      g1[0] = (int)(2u << 16);                     // data_size = 4 bytes
      g1[1] = (int)((uint32_t)(N_PTS & 0xffff) << 16);   // tensor_dim0 lo16
      g1[2] = (int)((N_PTS >> 16) | (3u << 16));   // dim0 hi16 | tensor_dim1 lo16
      g1[3] = (int)((uint32_t)CHUNK << 16);        // dim1 hi16=0 | tile_dim0
      g1[4] = 3;                                   // tile_dim1=3, tile_dim2=0
      g1[5] = N_PTS;                               // tensor_dim0_stride lo32
      g1[6] = 0;                                   // stride hi | dim1_stride lo
      g1[7] = 0;
      v4i gz = {0, 0, 0, 0};
#if __clang_major__ >= 23
      v8i gz8 = {0, 0, 0, 0, 0, 0, 0, 0};
      __builtin_amdgcn_tensor_load_to_lds(g0, g1, gz, gz, gz8, 0);
#else
      __builtin_amdgcn_tensor_load_to_lds(g0, g1, gz, gz, 0);
#endif
    }
    __builtin_amdgcn_s_wait_tensorcnt(0);
#elif defined(HAVE_ASYNC_LDS)
#pragma unroll
    for (int k = 0; k < 3 * CHUNK / (WG_THREADS * 4); ++k) {
      int vec = k * WG_THREADS + tid;
      int off = vec * 4;
      int c = off / CHUNK, p = off % CHUNK;
      float* src = const_cast<float*>(xb + c * N_PTS + cstart + p);
      __builtin_amdgcn_global_load_async_to_lds_b128(
          (gv4ip)src, (lv4ip)(&lds[off]), 0, 0);
    }
    wait_async_zero();
#else
#pragma unroll
    for (int k = 0; k < 3 * CHUNK / (WG_THREADS * 4); ++k) {
      int vec = k * WG_THREADS + tid;
      int off = vec * 4;
      int c = off / CHUNK, p = off % CHUNK;
      float4 v = *(const float4*)(xb + c * N_PTS + cstart + p);
      *(float4*)(&lds[off]) = v;
    }
#endif
    __syncthreads();

    // ---- scan 32 candidates per step ----
    for (int it = 0; it < CHUNK / 32; ++it) {
      const int base = it * 32;           // chunk-local candidate base
      const int cl   = base + lane;
      float c0 = lds[0 * CHUNK + cl];
      float c1 = lds[1 * CHUNK + cl];
      float c2 = lds[2 * CHUNK + cl];
      float nxx = -(c0 * c0 + c1 * c1 + c2 * c2);   // -(|c|^2), per-lane cand

      // B operands (4x16 f32), halves mirroring A; K=3 slot for lanes>=16
      // comes from the zero row (index select, no exec predication).
      const int nlo  = lane & 15;
      const int kk   = (lane < 16) ? 0 : 2;
      const int ybase = (lane < 16) ? (1 * CHUNK + base) : ZROW;
      const int yoff  = (lane < 16) ? 16 : 0;      // tile1 advance (zeros stay)
      v2f b0, b1;
      b0.x = lds[kk * CHUNK + base + nlo];
      b0.y = lds[ybase + nlo];
      b1.x = lds[kk * CHUNK + base + 16 + nlo];
      b1.y = lds[ybase + yoff + nlo];

      v8f cz = {0.f, 0.f, 0.f, 0.f, 0.f, 0.f, 0.f, 0.f};
      v8f d0 = __builtin_amdgcn_wmma_f32_16x16x4_f32(false, amat, false, b0,
                                                     (short)0, cz, false, false);
      v8f d1 = __builtin_amdgcn_wmma_f32_16x16x4_f32(false, amat, false, b1,
                                                     (short)0, cz, false, false);
      const int gbase = cstart + base;

      // Reassemble full 32-candidate rows from the two 16-wide D tiles.
#if __has_builtin(__builtin_amdgcn_permlane16_swap)
#define ROW_SWAP(E0, E1, RA, RB)                                               \
      {                                                                        \
        v2u _sw = __builtin_amdgcn_permlane16_swap(                            \
            __float_as_uint(E0), __float_as_uint(E1), false, false);           \
        RA = __uint_as_float(_sw[0]);   /* {E0.row0, E1.row0} */               \
        RB = __uint_as_float(_sw[1]);   /* {E0.row1, E1.row1} */               \
      }
#else
#define ROW_SWAP(E0, E1, RA, RB)                                               \
      {                                                                        \
        float _x0 = __shfl_xor((E0), 16);                                      \
        float _x1 = __shfl_xor((E1), 16);                                      \
        RA = (lane < 16) ? (E0) : _x1;                                         \
        RB = (lane < 16) ? _x0 : (E1);                                         \
      }
#endif

#define PROC_ROW(RR, ROWV)                                                     \
      {                                                                        \
        float sv = __builtin_fmaf((ROWV), 2.0f, nxx);  /* 2<q,c> - |c|^2 */    \
        float th = readlane_f32(res[RR].v, 0);         /* current 32nd */      \
        if (__any(sv > th)) {                                                  \
          VI nw; nw.v = sv; nw.i = gbase + lane;                               \
          res[RR] = topk_merge32(nw, res[RR]);                                 \
        }                                                                      \
      }

#pragma unroll
      for (int r8 = 0; r8 < 8; ++r8) {
        // D layout: vgpr r8 = rows {r8, r8+8}; lanes0-15 cols of tile.
        float rowA, rowB;
        ROW_SWAP(d0[r8], d1[r8], rowA, rowB);
        PROC_ROW(r8, rowA);        // row r8  , cands base+lane
        PROC_ROW(r8 + 8, rowB);    // row r8+8, cands base+lane
      }
#undef PROC_ROW
#undef ROW_SWAP
    }
    __syncthreads();   // before next refill overwrites LDS
  }

  // ---- resident lists sorted ascending by rank: lane31 = nearest.
  const int outRowBase = b * N_PTS + rowBase;
#pragma unroll
  for (int r = 0; r < 16; ++r)
    row_stats(xb, res[r].i, m8, m16, m32, outRowBase + r);
}

// Output assembly: out[b,f,c,n] via the closed-form of the two view-scrambles.
__global__ void assemble_kernel(const float* __restrict__ x,
                                const float* __restrict__ m8,
                                const float* __restrict__ m16,
                                const float* __restrict__ m32,
                                float* __restrict__ out) {
  const int N = N_PTS;
  int t = blockIdx.x * blockDim.x + threadIdx.x;
  if (t >= BATCH * 4 * 3 * N) return;
  int n = t % N;
  int rest = t / N;
  int c = rest % 3; rest /= 3;
  int f = rest % 4;
  int b = rest / 4;

  float val;
  if (f == 0) {
    val = m32[(b * N + n) * 3 + c];
  } else {
    int tt = 9 * n + 3 * (f - 1) + c;       // concat_x2 flat [B,3,N,3,1]
    int d  = tt / (3 * N);
    int rr = tt % (3 * N);
    int m  = rr / 3;
    int g  = rr % 3;
    if (g == 0) {
      val = m16[(b * N + m) * 3 + d];
    } else {
      int t2 = 6 * m + 3 * (g - 1) + d;     // concat_x1 flat [B,3,N,2,1]
      int d2 = t2 / (2 * N);
      int r2 = t2 % (2 * N);
      int m2 = r2 / 2;
      int g2 = r2 % 2;
      val = (g2 == 0) ? m8[(b * N + m2) * 3 + d2]
                      : x[((size_t)b * 3 + d2) * N + m2];
    }
  }
  out[t] = val;
}

extern "C" void kernel_launch(void* const* d_in, const int* in_sizes, int n_in,
                              void* d_out, int out_size, void* d_ws, size_t ws_size,
                              hipStream_t stream) {
  (void)in_sizes; (void)n_in; (void)ws_size; (void)out_size;
  const float* x = (const float*)d_in[0];
  float* ws  = (float*)d_ws;
  const size_t feat = (size_t)BATCH * N_PTS * 3;   // 98304 floats each
  float* m8  = ws;
  float* m16 = ws + feat;
  float* m32 = ws + 2 * feat;

  dim3 gridA(BATCH * (N_PTS / ROWS_PER_WG));       // 512 workgroups
  knn_feat_kernel<<<gridA, WG_THREADS, 0, stream>>>(x, m8, m16, m32);

  int total = BATCH * 4 * 3 * N_PTS;               // 393216 == out_size
  assemble_kernel<<<(total + 255) / 256, 256, 0, stream>>>(x, m8, m16, m32,
                                                           (float*)d_out);
}